// SupConLoss_88064009437247
// MI455X (gfx1250) — compile-verified
//
#include <hip/hip_runtime.h>
#include <math.h>

typedef __attribute__((ext_vector_type(2))) float v2f;
typedef __attribute__((ext_vector_type(8))) float v8f;

#define N 8192
#define D 512
#define INV_T 14.285714285714286f   // 1 / 0.07 == fixed softmax shift m0
#define EPSV 1e-12f
#define B_STRIDE 260                // 256 + 4 pad -> stride%64==4, conflict-free
#define NSLICE 8                    // column-tile slices (grid.y)
#define TILES_PER_SLICE (N / 16 / NSLICE)   // 64

// ---------------------------------------------------------------------------
// Kernel 1: L2-normalize each row. One wave per row (wave32), 8 rows/block.
// ---------------------------------------------------------------------------
__global__ __launch_bounds__(256) void supcon_normalize(
    const float* __restrict__ in, float* __restrict__ out) {
  const int wid  = threadIdx.x >> 5;
  const int lane = threadIdx.x & 31;
  const int row  = blockIdx.x * 8 + wid;
  const float* src = in + (size_t)row * D;
  float vals[16];
  float ss = 0.f;
#pragma unroll
  for (int i = 0; i < 16; ++i) {
    vals[i] = src[lane + i * 32];
    ss += vals[i] * vals[i];
  }
#pragma unroll
  for (int off = 16; off >= 1; off >>= 1) ss += __shfl_xor(ss, off, 32);
  const float r = rsqrtf(ss);
  float* dst = out + (size_t)row * D;
#pragma unroll
  for (int i = 0; i < 16; ++i) dst[lane + i * 32] = vals[i] * r;
}

// Cooperative copy of one 16-row x 256-col half-tile of B into padded LDS.
__device__ __forceinline__ void stage_half_tile(
    const float* __restrict__ F, float* __restrict__ dst, int tid,
    int tile, int half) {
  const float* src = F + (size_t)(tile * 16) * D + half * 256;
#pragma unroll
  for (int i = 0; i < 4; ++i) {
    const int idx4  = tid + i * 256;  // 0..1023 float4s
    const int brow  = idx4 >> 6;      // 64 float4 per row
    const int bcol4 = idx4 & 63;
    const float4 v = *reinterpret_cast<const float4*>(
        src + (size_t)brow * D + bcol4 * 4);
    *reinterpret_cast<float4*>(dst + brow * B_STRIDE + bcol4 * 4) = v;
  }
}

// 64 WMMAs on one K=256 half-tile, with a software-pipelined B stream:
// 8 v2f B-registers in flight so ds_load latency never gates WMMA issue.
// AOFF/BUF are literal constants -> a[] stays VGPR-resident after unroll.
#define COMPUTE_HALF(AOFF, BUF)                                              \
  {                                                                          \
    const float* Bb = &Bs[BUF][lh * B_STRIDE + khalf];                       \
    v2f bcur[8], bnxt[8];                                                    \
    _Pragma("unroll")                                                        \
    for (int p = 0; p < 8; ++p)                                              \
      bcur[p] = *reinterpret_cast<const v2f*>(Bb + 4 * p);                   \
    _Pragma("unroll")                                                        \
    for (int g = 0; g < 8; ++g) {                                            \
      if (g < 7) {                                                           \
        _Pragma("unroll")                                                    \
        for (int p = 0; p < 8; ++p)                                          \
          bnxt[p] =                                                          \
              *reinterpret_cast<const v2f*>(Bb + 4 * ((g + 1) * 8 + p));     \
      }                                                                      \
      _Pragma("unroll")                                                      \
      for (int p = 0; p < 8; ++p)                                            \
        c = __builtin_amdgcn_wmma_f32_16x16x4_f32(                           \
            false, a[(AOFF) + g * 8 + p], false, bcur[p], (short)0, c,       \
            false, false);                                                   \
      _Pragma("unroll")                                                      \
      for (int p = 0; p < 8; ++p) bcur[p] = bnxt[p];                         \
    }                                                                        \
  }

// ---------------------------------------------------------------------------
// Kernel 2: fused Gram-matrix (fp32 WMMA) + fixed-shift softmax statistics.
// Grid (64, 8): blockIdx.x -> 128-row block (8 waves x 16 rows, A-tile held
// in VGPRs per wave), blockIdx.y -> slice of 64 column tiles. The 16-col B
// tile is shared by all 8 waves via double-buffered LDS half-tiles (K=256),
// staged ahead of the 64-WMMA compute block to overlap load latency.
// Per row: s = sum exp(v - 1/T), sml = sum(mask*v), cnt = sum(mask),
//          selfv = v_ii. Partials written to ws; merged by rowfinish kernel.
// ---------------------------------------------------------------------------
__global__ __launch_bounds__(256, 1) void supcon_main(
    const float* __restrict__ F, const int* __restrict__ labels,
    float* __restrict__ P) {
  __shared__ __align__(16) float Bs[2][16 * B_STRIDE];

  const int tid     = threadIdx.x;
  const int wid     = tid >> 5;
  const int lane    = tid & 31;
  const int lh      = lane & 15;          // position within 16-lane half
  const int khalf   = (lane >> 4) << 1;   // K offset 0 or 2 (WMMA f32 layout)
  const int rowHalf = (lane >> 4) << 3;   // C rows: r (half0) / r+8 (half1)
  const int rowBase = blockIdx.x * 128 + wid * 16;
  const int jBase   = blockIdx.y * TILES_PER_SLICE;

  // A tile: 16 rows x 512 K, register-resident (256 VGPRs, gfx1250 big RF).
  v2f a[128];
  const float* Arow = F + (size_t)(rowBase + lh) * D + khalf;
#pragma unroll
  for (int u = 0; u < 128; ++u)
    a[u] = *reinterpret_cast<const v2f*>(Arow + 4 * u);

  int rowLab[8];
#pragma unroll
  for (int r = 0; r < 8; ++r) rowLab[r] = labels[rowBase + rowHalf + r];

  float s[8], sml[8], cnt[8], selfv[8];
#pragma unroll
  for (int r = 0; r < 8; ++r) {
    s[r] = 0.f; sml[r] = 0.f; cnt[r] = 0.f; selfv[r] = -__builtin_inff();
  }

  stage_half_tile(F, Bs[0], tid, jBase, 0);
  __syncthreads();

  for (int t = 0; t < TILES_PER_SLICE; ++t) {
    const int jj     = jBase + t;
    const int gcol   = jj * 16 + lh;
    const int colLab = labels[gcol];   // issued early, consumed in epilogue

    v8f c = {};

    // ---- stage (jj, half1) while computing half0 from Bs[0] ----
    stage_half_tile(F, Bs[1], tid, jj, 1);
    COMPUTE_HALF(0, 0)
    __syncthreads();  // Bs[1] staged; Bs[0] reads retired

    // ---- stage (jj+1, half0) while computing half1 from Bs[1] ----
    if (t + 1 < TILES_PER_SLICE) stage_half_tile(F, Bs[0], tid, jj + 1, 0);
    COMPUTE_HALF(64, 1)

    // ---- epilogue: fold the finished 16x16 logit tile into row stats ----
#pragma unroll
    for (int r = 0; r < 8; ++r) {
      const float v   = c[r] * INV_T;
      const float msk = (rowLab[r] == colLab) ? 1.f : 0.f;
      sml[r] += msk * v;
      cnt[r] += msk;
      if (rowBase + rowHalf + r == gcol) selfv[r] = v;   // diagonal logit
      s[r] += __expf(v - INV_T);   // bounded: logit <= 1/T (normalized rows)
    }
    __syncthreads();  // Bs[0] staged; Bs[1] reads retired
  }

  // Merge partials across the 16 lanes of each half (wave32 -> 4 xor steps).
#pragma unroll
  for (int off = 1; off < 16; off <<= 1) {
#pragma unroll
    for (int r = 0; r < 8; ++r) {
      s[r]     += __shfl_xor(s[r], off, 32);
      sml[r]   += __shfl_xor(sml[r], off, 32);
      cnt[r]   += __shfl_xor(cnt[r], off, 32);
      selfv[r]  = fmaxf(selfv[r], __shfl_xor(selfv[r], off, 32));
    }
  }

  // Lanes 0 and 16 hold full slice-partials for rows rowHalf..rowHalf+7.
  if (lh == 0) {
    float* base = P + (size_t)blockIdx.y * N;   // slice-major layout
#pragma unroll
    for (int r = 0; r < 8; ++r) {
      const int row = rowBase + rowHalf + r;
      base[row]                          = s[r];
      base[(size_t)NSLICE * N + row]     = sml[r];
      base[(size_t)2 * NSLICE * N + row] = cnt[r];
      base[(size_t)3 * NSLICE * N + row] = selfv[r];
    }
  }
}

// ---------------------------------------------------------------------------
// Kernel 3: merge the 8 column-slices per row, finish the per-row loss,
// block-reduce. Deterministic (no atomics).
// ---------------------------------------------------------------------------
__global__ __launch_bounds__(256) void supcon_rowfinish(
    const float* __restrict__ P, float* __restrict__ partials) {
  const int row = blockIdx.x * 256 + threadIdx.x;
  float S = 0.f, L = 0.f, C2 = 0.f, SF = -__builtin_inff();
#pragma unroll
  for (int sl = 0; sl < NSLICE; ++sl) {
    const float* base = P + (size_t)sl * N;
    S  += base[row];
    L  += base[(size_t)NSLICE * N + row];
    C2 += base[(size_t)2 * NSLICE * N + row];
    SF  = fmaxf(SF, base[(size_t)3 * NSLICE * N + row]);
  }
  const float denom = S - __expf(SF - INV_T);          // drop self-similarity
  float mlpp = L / C2 - INV_T - __logf(denom + EPSV);  // row-max cancels

  __shared__ float sm[256];
  sm[threadIdx.x] = mlpp;
  __syncthreads();
  for (int off = 128; off >= 32; off >>= 1) {
    if (threadIdx.x < off) sm[threadIdx.x] += sm[threadIdx.x + off];
    __syncthreads();
  }
  if (threadIdx.x < 32) {
    float x = sm[threadIdx.x];
#pragma unroll
    for (int off = 16; off >= 1; off >>= 1) x += __shfl_xor(x, off, 32);
    if (threadIdx.x == 0) partials[blockIdx.x] = x;
  }
}

// ---------------------------------------------------------------------------
// Kernel 4: final reduction of 32 block partials -> loss.
// ---------------------------------------------------------------------------
__global__ void supcon_reduce(const float* __restrict__ partials,
                              float* __restrict__ out) {
  float v = partials[threadIdx.x];
#pragma unroll
  for (int off = 16; off >= 1; off >>= 1) v += __shfl_xor(v, off, 32);
  if (threadIdx.x == 0) out[0] = -v / (float)N;
}

extern "C" void kernel_launch(void* const* d_in, const int* in_sizes, int n_in,
                              void* d_out, int out_size, void* d_ws, size_t ws_size,
                              hipStream_t stream) {
  (void)in_sizes; (void)n_in; (void)out_size; (void)ws_size;
  const float* features = (const float*)d_in[0];
  const int*   labels   = (const int*)d_in[1];
  float* out      = (float*)d_out;
  float* normF    = (float*)d_ws;                       // 8192*512 f = 16 MB
  float* P        = normF + (size_t)N * D;              // 4*8*8192 f = 1 MB
  float* partials = P + (size_t)4 * NSLICE * N;         // 32 floats

  supcon_normalize<<<N / 8, 256, 0, stream>>>(features, normF);
  supcon_main<<<dim3(N / 128, NSLICE), 256, 0, stream>>>(normF, labels, P);
  supcon_rowfinish<<<N / 256, 256, 0, stream>>>(P, partials);
  supcon_reduce<<<1, 32, 0, stream>>>(partials, out);
}